// Edge_Labelling_54348516164305
// MI455X (gfx1250) — compile-verified
//
#include <hip/hip_runtime.h>
#include <hip/hip_bf16.h>
#include <math.h>

typedef __attribute__((ext_vector_type(2))) float v2f;
typedef __attribute__((ext_vector_type(8))) float v8f;

#define DIM 128

// ---------------------------------------------------------------------------
// Kernel 1: fold W2/b1/b2 through W1.
//   v1 = W1^T (w_a + w_b), v2 = W1^T (w_b - w_a), C = 2*(b1.w_b) + b2
// where w_a = W2[0,0:128] (diff weights), w_b = W2[0,128:256] (sum weights).
// One block of 128 threads; negligible cost.
// ---------------------------------------------------------------------------
__global__ void prep_fold_kernel(const float* __restrict__ W1,
                                 const float* __restrict__ b1,
                                 const float* __restrict__ W2,
                                 const float* __restrict__ b2,
                                 float* __restrict__ v1,
                                 float* __restrict__ v2,
                                 float* __restrict__ Cout) {
    int d = threadIdx.x;  // 0..127
    float s1 = 0.0f, s2 = 0.0f;
    for (int o = 0; o < DIM; ++o) {
        float wa = W2[o];         // weight on (h_src - h_dst)
        float wb = W2[DIM + o];   // weight on (h_src + h_dst)
        float w  = W1[o * DIM + d];
        s1 += (wa + wb) * w;
        s2 += (wb - wa) * w;
    }
    v1[d] = s1;
    v2[d] = s2;
    if (d == 0) {
        float c = b2[0];
        for (int o = 0; o < DIM; ++o) c += 2.0f * b1[o] * W2[DIM + o];
        *Cout = c;
    }
}

// ---------------------------------------------------------------------------
// Kernel 2: per-node scalars via f32 WMMA (exact f32 accumulate).
//   pq[2n+0] = f[n].v1 + C/2,  pq[2n+1] = f[n].v2 + C/2
// One wave32 per 16-row tile. B = [v1 | v2 | 0...] zero-padded to 128x16,
// staged column-major in LDS so the inner loop is branch-free:
//   global_load_b64 (A frag) + ds_load_2addr_b64 (B frags) + v_wmma_f32_16x16x4_f32.
// A 16x4 f32 layout: VGPR0 = K+0 (lanes 0-15) / K+2 (lanes 16-31), VGPR1 = K+1/K+3.
// C/D 16x16 f32 layout: VGPR r -> M=r (lanes 0-15), M=8+r (lanes 16-31), N=lane%16.
// ---------------------------------------------------------------------------
__global__ void node_pq_wmma_kernel(const float* __restrict__ F,
                                    const float* __restrict__ v1g,
                                    const float* __restrict__ v2g,
                                    const float* __restrict__ Cptr,
                                    float* __restrict__ pq,
                                    int N, int numTiles) {
    // Zero-padded B, column-major: sB[n*DIM + k], n = 0..15 (8 KB of 320 KB WGP LDS)
    __shared__ float sB[16 * DIM];
    int tid = threadIdx.x;  // blockDim.x == 256
    for (int i = tid; i < 16 * DIM; i += 256) {
        int n = i >> 7;          // column
        int k = i & (DIM - 1);   // row (K index)
        float v = 0.0f;
        if (n == 0) v = v1g[k];
        else if (n == 1) v = v2g[k];
        sB[i] = v;
    }
    __syncthreads();

    int wave = tid >> 5;           // 8 waves per block, one 16-row tile each
    int lane = tid & 31;
    int tile = blockIdx.x * 8 + wave;
    if (tile >= numTiles) return;  // wave-uniform: EXEC all-ones for WMMA

    int half   = lane >> 4;        // 0 -> K+0/K+1 ; 1 -> K+2/K+3
    int lane_n = lane & 15;
    int rowA   = tile * 16 + lane_n;
    if (rowA >= N) rowA = N - 1;   // clamp (stores guarded); no conditional loads
    const float* frow = F  + (size_t)rowA * DIM + half * 2;
    const float* bcol = sB + lane_n * DIM       + half * 2;

    v8f acc = {};
    #pragma unroll 8
    for (int k0 = 0; k0 < DIM; k0 += 4) {
        v2f a = *(const v2f*)(frow + k0);   // global_load_b64, unconditional
        v2f b = *(const v2f*)(bcol + k0);   // ds_load_b64, unconditional
        acc = __builtin_amdgcn_wmma_f32_16x16x4_f32(
            /*neg_a=*/false, a, /*neg_b=*/false, b,
            /*c_mod=*/(short)0, acc, /*reuse_a=*/false, /*reuse_b=*/false);
    }

    float halfC = 0.5f * (*Cptr);  // fold edge-constant into both p and q
    if (lane_n < 2) {
        int base = tile * 16 + half * 8;
        if (tile * 16 + 16 <= N) {
            // wave-uniform fast path (every tile when N % 16 == 0): no guards
            #pragma unroll
            for (int r = 0; r < 8; ++r)
                pq[2 * (base + r) + lane_n] = acc[r] + halfC;
        } else {
            // tail tile only
            #pragma unroll
            for (int r = 0; r < 8; ++r) {
                int row = base + r;
                if (row < N) pq[2 * row + lane_n] = acc[r] + halfC;
            }
        }
    }
}

// ---------------------------------------------------------------------------
// Kernel 3: per-edge gather + tanh. pq table (~400 KB) is L2-resident;
// out[e] = tanh(p'[src] + q'[dst]) with C pre-folded. Pure streaming kernel.
// ---------------------------------------------------------------------------
__global__ void edge_tanh_kernel(const int* __restrict__ src,
                                 const int* __restrict__ dst,
                                 const float* __restrict__ pq,
                                 float* __restrict__ out, int E) {
    int e = blockIdx.x * blockDim.x + threadIdx.x;
    if (e >= E) return;
    int s = src[e];
    int d = dst[e];
    out[e] = tanhf(pq[2 * s] + pq[2 * d + 1]);
}

// ---------------------------------------------------------------------------
// Launch: prep -> node WMMA GEMM -> edge pass, ordered on `stream`.
// Workspace: [v1:128][v2:128][C:1][pad][pq:2*N] floats (~401 KB total).
// ---------------------------------------------------------------------------
extern "C" void kernel_launch(void* const* d_in, const int* in_sizes, int n_in,
                              void* d_out, int out_size, void* d_ws, size_t ws_size,
                              hipStream_t stream) {
    const float* features = (const float*)d_in[0];
    const int*   src      = (const int*)d_in[1];
    const int*   dst      = (const int*)d_in[2];
    const float* W1       = (const float*)d_in[3];
    const float* b1       = (const float*)d_in[4];
    const float* W2       = (const float*)d_in[5];
    const float* b2       = (const float*)d_in[6];
    float* out = (float*)d_out;

    int N = in_sizes[0] / DIM;   // 50000
    int E = in_sizes[1];         // 800000

    float* ws = (float*)d_ws;
    float* v1 = ws;
    float* v2 = ws + DIM;
    float* Cv = ws + 2 * DIM;
    float* pq = ws + 2 * DIM + 64;  // aligned scratch for 2*N floats

    prep_fold_kernel<<<1, DIM, 0, stream>>>(W1, b1, W2, b2, v1, v2, Cv);

    int numTiles = (N + 15) / 16;
    int blocks   = (numTiles + 7) / 8;  // 8 waves (tiles) per 256-thread block
    node_pq_wmma_kernel<<<blocks, 256, 0, stream>>>(features, v1, v2, Cv, pq, N, numTiles);

    edge_tanh_kernel<<<(E + 255) / 256, 256, 0, stream>>>(src, dst, pq, out, E);
}